// HeteroBlock_44341242364503
// MI455X (gfx1250) — compile-verified
//
#include <hip/hip_runtime.h>
#include <math.h>

typedef float v2f __attribute__((ext_vector_type(2)));
typedef float v8f __attribute__((ext_vector_type(8)));

#define NG 50000
#define NL 200000
#define NC 1000
#define DD 128

// ---------------------------------------------------------------------------
// sum of up to 3 arrays (for ΣWr, Σbias per destination type)
// ---------------------------------------------------------------------------
__global__ void sum3_kernel(const float* __restrict__ a, const float* __restrict__ b,
                            const float* __restrict__ c, float* __restrict__ o, int n) {
    int i = blockIdx.x * blockDim.x + threadIdx.x;
    if (i < n) {
        float v = a[i] + b[i];
        if (c) v += c[i];
        o[i] = v;
    }
}

// ---------------------------------------------------------------------------
// edge aggregation: one wave32 per edge; lane covers 4 contiguous floats.
// Coalesced 512B gather per edge, f32 atomics into agg (L2-resident).
// ---------------------------------------------------------------------------
__global__ __launch_bounds__(256) void edge_agg_kernel(
    const float* __restrict__ xsrc, const int* __restrict__ src,
    const int* __restrict__ dst, float* __restrict__ agg,
    float* __restrict__ cnt, int nE) {
    int gt = blockIdx.x * blockDim.x + threadIdx.x;
    int wave = gt >> 5;
    int lane = threadIdx.x & 31;
    int nWaves = (gridDim.x * blockDim.x) >> 5;
    for (int e = wave; e < nE; e += nWaves) {
        int s = src[e];
        int d = dst[e];
        const float4 v = *(const float4*)(xsrc + (size_t)s * DD + lane * 4);
        float* ag = agg + (size_t)d * DD + lane * 4;
        atomicAdd(ag + 0, v.x);
        atomicAdd(ag + 1, v.y);
        atomicAdd(ag + 2, v.z);
        atomicAdd(ag + 3, v.w);
        if (lane == 0) atomicAdd(cnt + d, 1.0f);
    }
}

// ---------------------------------------------------------------------------
// Fused: [mean_et @ Wl_et (WMMA f32)] + x @ ΣWr (WMMA f32) + Σbias
//        -> exact GELU -> +x residual -> LayerNorm -> out
// 256 threads = 8 wave32; each wave computes a 16x128 output tile via
// v_wmma_f32_16x16x4_f32. K streamed through LDS in 32-wide panels staged
// with GLOBAL_LOAD_ASYNC_TO_LDS (ASYNCcnt path) — no VGPR round trip.
// ---------------------------------------------------------------------------
#define KP 32          // K panel width
#define KPAD 36        // padded row stride (floats) -> conflict-free b64 frag loads
__global__ __launch_bounds__(256) void fused_sage_kernel(
    const float* __restrict__ xdst, int N,
    const float* __restrict__ wr_sum,   // [128][128] row-major (k,n)
    const float* __restrict__ bsum,     // [128]
    const float* a0, const float* c0, const float* w0,
    const float* a1, const float* c1, const float* w1,
    const float* a2, const float* c2, const float* w2,
    int nsrc,
    const float* __restrict__ gamma, const float* __restrict__ beta,
    float* __restrict__ out) {

    __shared__ __align__(16) float ldsW[128][KPAD];   // weight panel, transposed [n][k_local]
    __shared__ __align__(16) float ldsA[8][16][KPAD]; // per-wave A panel [row][k_local]

    const int tid  = threadIdx.x;
    const int w    = tid >> 5;
    const int lane = tid & 31;
    const int p16  = lane & 15;
    const int hi   = lane >> 4;            // half-wave select
    const int rowBase = blockIdx.x * 128 + w * 16;

    v8f acc[8];
#pragma unroll
    for (int j = 0; j < 8; ++j) acc[j] = (v8f)0.0f;

    const float* mats[4] = {xdst, a0, a1, a2};
    const float* cnts[4] = {nullptr, c0, c1, c2};
    const float* wms[4]  = {wr_sum, w0, w1, w2};
    const int ns = nsrc + 1;

    // A-panel async copy geometry: 4 ops x (4 rows, 8 lanes/row, 16B/lane)
    const int rsub = lane >> 3;            // row within quad
    const unsigned csub = (unsigned)(lane & 7) * 16u;  // byte offset in 128B row

    for (int s = 0; s < ns; ++s) {
        const float* M = mats[s];
        const float* C = cnts[s];
        const float* W = wms[s];
        // row-wise mean scale folded into the A fragment (row = p16 of this tile)
        float scaleA = 1.0f;
        if (C) {
            int srow = rowBase + p16;
            float c = (srow < N) ? C[srow] : 1.0f;
            scaleA = 1.0f / fmaxf(c, 1.0f);
        }
        for (int p = 0; p < 4; ++p) {
            __syncthreads();               // protect previous panel reads
            const int k0 = p * KP;
            // ---- stage weight panel transposed via async b32 scatter to LDS ----
#pragma unroll
            for (int it = 0; it < 16; ++it) {
                int i = tid + it * 256;
                int kl = i >> 7;
                int n  = i & 127;
                unsigned ldsAddr = (unsigned)(size_t)&ldsW[n][kl];
                unsigned goff = ((unsigned)(k0 + kl) * DD + (unsigned)n) * 4u;
                asm volatile("global_load_async_to_lds_b32 %0, %1, %2"
                             :: "v"(ldsAddr), "v"(goff), "s"(W) : "memory");
            }
            // ---- stage this wave's 16x32 A panel via async b128 copies ----
#pragma unroll
            for (int q = 0; q < 4; ++q) {
                int r = q * 4 + rsub;
                int row = rowBase + r;
                if (row >= N) row = N - 1;     // clamp: garbage rows never stored
                unsigned ldsAddr = (unsigned)(size_t)&ldsA[w][r][0] + csub;
                unsigned goff = ((unsigned)row * DD + (unsigned)k0) * 4u + csub;
                asm volatile("global_load_async_to_lds_b128 %0, %1, %2"
                             :: "v"(ldsAddr), "v"(goff), "s"(M) : "memory");
            }
            asm volatile("s_wait_asynccnt 0x0" ::: "memory");
            __syncthreads();
            // ---- 8 K-steps of 4, 8 N-tiles -> 64 WMMAs per panel ----
#pragma unroll
            for (int kk = 0; kk < 8; ++kk) {
                const int kloc = kk * 4 + hi * 2;
                v2f a = *(const v2f*)&ldsA[w][p16][kloc];   // A frag: M=p16, K=kloc+{0,1}
                a *= scaleA;                                // fold mean divide
#pragma unroll
                for (int j = 0; j < 8; ++j) {
                    v2f b = *(const v2f*)&ldsW[16 * j + p16][kloc];  // B frag
                    acc[j] = __builtin_amdgcn_wmma_f32_16x16x4_f32(
                        false, a, false, b, (short)0, acc[j], false, false);
                }
            }
        }
    }

    // ---- epilogue: bias + GELU + residual + LayerNorm, all in registers ----
    // C-frag layout: acc[j][v] = element (row = v + 8*hi, col = 16*j + p16)
    float sum1[8];
#pragma unroll
    for (int v = 0; v < 8; ++v) sum1[v] = 0.0f;

#pragma unroll
    for (int j = 0; j < 8; ++j) {
        const int col = 16 * j + p16;
        const float bv = bsum[col];
#pragma unroll
        for (int v = 0; v < 8; ++v) {
            const int row = rowBase + v + 8 * hi;
            float h = acc[j][v] + bv;
            float g = 0.5f * h * (1.0f + erff(h * 0.70710678118654752f)); // exact GELU
            float xr = (row < N) ? xdst[(size_t)row * DD + col] : 0.0f;
            float y = g + xr;
            acc[j][v] = y;
            sum1[v] += y;
        }
    }
    // reduce across the 16 lanes holding each row (masks stay within half-wave)
    float mu[8];
#pragma unroll
    for (int v = 0; v < 8; ++v) {
        float s = sum1[v];
        s += __shfl_xor(s, 1);
        s += __shfl_xor(s, 2);
        s += __shfl_xor(s, 4);
        s += __shfl_xor(s, 8);
        mu[v] = s * (1.0f / 128.0f);
    }
    float sum2[8];
#pragma unroll
    for (int v = 0; v < 8; ++v) sum2[v] = 0.0f;
#pragma unroll
    for (int j = 0; j < 8; ++j)
#pragma unroll
        for (int v = 0; v < 8; ++v) {
            float d = acc[j][v] - mu[v];
            sum2[v] += d * d;
        }
    float rs[8];
#pragma unroll
    for (int v = 0; v < 8; ++v) {
        float s = sum2[v];
        s += __shfl_xor(s, 1);
        s += __shfl_xor(s, 2);
        s += __shfl_xor(s, 4);
        s += __shfl_xor(s, 8);
        rs[v] = rsqrtf(s * (1.0f / 128.0f) + 1e-5f);
    }
#pragma unroll
    for (int j = 0; j < 8; ++j) {
        const int col = 16 * j + p16;
        const float gm = gamma[col];
        const float bt = beta[col];
#pragma unroll
        for (int v = 0; v < 8; ++v) {
            const int row = rowBase + v + 8 * hi;
            if (row < N)
                out[(size_t)row * DD + col] = (acc[j][v] - mu[v]) * rs[v] * gm + bt;
        }
    }
}

// ---------------------------------------------------------------------------
extern "C" void kernel_launch(void* const* d_in, const int* in_sizes, int n_in,
                              void* d_out, int out_size, void* d_ws, size_t ws_size,
                              hipStream_t stream) {
    const float* x_g = (const float*)d_in[0];
    const float* x_l = (const float*)d_in[1];
    const float* x_c = (const float*)d_in[2];
    // per edge type et (gl,lg,ll,lc,cl,gc,cg): base=3+5*i : src,dst,Wl,bl,Wr
    const int* e_src[7];
    const int* e_dst[7];
    const float* Wl[7];
    const float* bl[7];
    const float* Wr[7];
    int nE[7];
    for (int i = 0; i < 7; ++i) {
        int b = 3 + 5 * i;
        e_src[i] = (const int*)d_in[b + 0];
        e_dst[i] = (const int*)d_in[b + 1];
        Wl[i]    = (const float*)d_in[b + 2];
        bl[i]    = (const float*)d_in[b + 3];
        Wr[i]    = (const float*)d_in[b + 4];
        nE[i]    = in_sizes[b + 0];
    }
    enum { GL = 0, LG = 1, LL = 2, LC = 3, CL = 4, GC = 5, CG = 6 };
    const float* g_g = (const float*)d_in[38];
    const float* b_g = (const float*)d_in[39];
    const float* g_l = (const float*)d_in[40];
    const float* b_l = (const float*)d_in[41];
    const float* g_c = (const float*)d_in[42];
    const float* b_c = (const float*)d_in[43];

    // workspace layout
    float* ws = (float*)d_ws;
    size_t o = 0;
    float* agg_gl = ws + o; o += (size_t)NL * DD;
    float* agg_ll = ws + o; o += (size_t)NL * DD;
    float* agg_cl = ws + o; o += (size_t)NL * DD;
    float* agg_lg = ws + o; o += (size_t)NG * DD;
    float* agg_cg = ws + o; o += (size_t)NG * DD;
    float* agg_lc = ws + o; o += (size_t)NC * DD;
    float* agg_gc = ws + o; o += (size_t)NC * DD;
    float* cnt_gl = ws + o; o += NL;
    float* cnt_ll = ws + o; o += NL;
    float* cnt_cl = ws + o; o += NL;
    float* cnt_lg = ws + o; o += NG;
    float* cnt_cg = ws + o; o += NG;
    float* cnt_lc = ws + o; o += NC;
    float* cnt_gc = ws + o; o += NC;
    const size_t zeroBytes = o * sizeof(float);
    float* wrs_g = ws + o; o += DD * DD;
    float* wrs_l = ws + o; o += DD * DD;
    float* wrs_c = ws + o; o += DD * DD;
    float* bs_g  = ws + o; o += DD;
    float* bs_l  = ws + o; o += DD;
    float* bs_c  = ws + o; o += DD;

    float* out_g = (float*)d_out;
    float* out_l = out_g + (size_t)NG * DD;
    float* out_c = out_l + (size_t)NL * DD;

    hipMemsetAsync(d_ws, 0, zeroBytes, stream);

    // ΣWr / Σbias per destination type
    sum3_kernel<<<(DD * DD + 255) / 256, 256, 0, stream>>>(Wr[LG], Wr[CG], nullptr, wrs_g, DD * DD);
    sum3_kernel<<<(DD * DD + 255) / 256, 256, 0, stream>>>(Wr[GL], Wr[LL], Wr[CL], wrs_l, DD * DD);
    sum3_kernel<<<(DD * DD + 255) / 256, 256, 0, stream>>>(Wr[LC], Wr[GC], nullptr, wrs_c, DD * DD);
    sum3_kernel<<<1, 256, 0, stream>>>(bl[LG], bl[CG], nullptr, bs_g, DD);
    sum3_kernel<<<1, 256, 0, stream>>>(bl[GL], bl[LL], bl[CL], bs_l, DD);
    sum3_kernel<<<1, 256, 0, stream>>>(bl[LC], bl[GC], nullptr, bs_c, DD);

    // scatter-mean aggregation per edge type
    const int EB = 4096;
    edge_agg_kernel<<<EB, 256, 0, stream>>>(x_g, e_src[GL], e_dst[GL], agg_gl, cnt_gl, nE[GL]);
    edge_agg_kernel<<<EB, 256, 0, stream>>>(x_l, e_src[LG], e_dst[LG], agg_lg, cnt_lg, nE[LG]);
    edge_agg_kernel<<<EB, 256, 0, stream>>>(x_l, e_src[LL], e_dst[LL], agg_ll, cnt_ll, nE[LL]);
    edge_agg_kernel<<<EB, 256, 0, stream>>>(x_l, e_src[LC], e_dst[LC], agg_lc, cnt_lc, nE[LC]);
    edge_agg_kernel<<<EB, 256, 0, stream>>>(x_c, e_src[CL], e_dst[CL], agg_cl, cnt_cl, nE[CL]);
    edge_agg_kernel<<<EB, 256, 0, stream>>>(x_g, e_src[GC], e_dst[GC], agg_gc, cnt_gc, nE[GC]);
    edge_agg_kernel<<<EB, 256, 0, stream>>>(x_c, e_src[CG], e_dst[CG], agg_cg, cnt_cg, nE[CG]);

    // fused WMMA GEMM + GELU + residual + LayerNorm per destination type
    fused_sage_kernel<<<(NG + 127) / 128, 256, 0, stream>>>(
        x_g, NG, wrs_g, bs_g,
        agg_lg, cnt_lg, Wl[LG], agg_cg, cnt_cg, Wl[CG], nullptr, nullptr, nullptr, 2,
        g_g, b_g, out_g);
    fused_sage_kernel<<<(NL + 127) / 128, 256, 0, stream>>>(
        x_l, NL, wrs_l, bs_l,
        agg_gl, cnt_gl, Wl[GL], agg_ll, cnt_ll, Wl[LL], agg_cl, cnt_cl, Wl[CL], 3,
        g_l, b_l, out_l);
    fused_sage_kernel<<<(NC + 127) / 128, 256, 0, stream>>>(
        x_c, NC, wrs_c, bs_c,
        agg_lc, cnt_lc, Wl[LC], agg_gc, cnt_gc, Wl[GC], nullptr, nullptr, nullptr, 2,
        g_c, b_c, out_c);
}